// L1DifferenceLayer_27212912788354
// MI455X (gfx1250) — compile-verified
//
#include <hip/hip_runtime.h>
#include <hip/hip_bf16.h>

// Problem constants (from reference)
#define N_NODES 50000
#define N_EDGES 800000
#define N_FEAT  16
#define N_BASIS 4
#define GAMMA_C 10.0f
#define CENTER_STEP (5.0f / 3.0f)      // linspace(0, 5, 4) step
#define INV_SQRT2_C 0.70710678118654752440f
#define ACC_STRIDE (3 * N_BASIS * N_FEAT)   // 192 floats per node

typedef __attribute__((ext_vector_type(2))) float v2f;
typedef __attribute__((ext_vector_type(8))) float v8f;

// ---------------------------------------------------------------------------
// Kernel 1: zero the accumulator (imag half of d_out, N_NODES*192 floats).
// segment_sum must produce zeros for nodes with no incident edges.
// ---------------------------------------------------------------------------
__global__ void zero_acc_kernel(float* __restrict__ acc, int n) {
    int i = blockIdx.x * blockDim.x + threadIdx.x;
    if (i < n) acc[i] = 0.0f;
}

// ---------------------------------------------------------------------------
// Kernel 2: one wave per edge.
//   u[m] = unit[m/4] * exp(-10*(dist - center[m%4])^2)   (m = 0..11)
//   D(16x16) = A(16x4) * B(4x16), A col0 = u (zero-padded), B row0 = nf(dst)
//   Then 8 coalesced global_atomic_add_f32 rows into acc[src].
// All lanes compute the scalar edge quantities redundantly (cheap, no LDS).
// Operand construction is branch-free (cndmask selects, clamped-lane load).
// ---------------------------------------------------------------------------
__global__ __launch_bounds__(256)
void edge_outer_wmma_kernel(const float* __restrict__ pos,
                            const float* __restrict__ node_features,
                            const int*   __restrict__ edge_idx,
                            float*       __restrict__ acc) {
    const int wave = blockIdx.x * 8 + (threadIdx.x >> 5);   // grid sized exactly
    const int lane = threadIdx.x & 31;

    const int s = edge_idx[wave];             // src
    const int d = edge_idx[N_EDGES + wave];   // dst

    // Unconditional, coalesced: lanes 16-31 re-read the same 16 floats (L0 hit).
    const float bload = node_features[N_FEAT * d + (lane & 15)];

    const float dx = pos[3 * d + 0] - pos[3 * s + 0];
    const float dy = pos[3 * d + 1] - pos[3 * s + 1];
    const float dz = pos[3 * d + 2] - pos[3 * s + 2];
    const float dist = sqrtf(dx * dx + dy * dy + dz * dz);
    const float rinv = 1.0f / dist;

    // A matrix (16x4 f32): lanes 0-15 hold {K=0,K=1}, lanes 16-31 hold {K=2,K=3}.
    // Only (M<12, K=0) is nonzero: u[m] = unit[m/4] * bf[m%4]. Branch-free:
    // compute on all lanes (clamped), zero out lanes >= 12 with a select.
    const int c = lane >> 2;                       // 0..7; only 0..2 survive
    const float comp = (c == 0) ? dx : ((c == 1) ? dy : dz);
    const float t = dist - CENTER_STEP * (float)(lane & 3);
    const float u = (comp * rinv) * expf(-GAMMA_C * t * t);
    const float a0 = (lane < 12) ? u : 0.0f;

    // B matrix (4x16 f32): only row K=0 nonzero: nf[dst][n] on lanes 0-15, VGPR0.
    const float b0 = (lane < 16) ? bload : 0.0f;

    v2f A; A.x = a0; A.y = 0.0f;
    v2f B; B.x = b0; B.y = 0.0f;
    v8f C = {};

    v8f D = __builtin_amdgcn_wmma_f32_16x16x4_f32(
        /*neg_a=*/false, A, /*neg_b=*/false, B,
        /*c_mod=*/(short)0, C, /*reuse_a=*/false, /*reuse_b=*/false);

    // D layout: lane L -> col N = L%16; VGPR j -> row M = j + (L<16 ? 0 : 8).
    // acc offset for row M, col f: c*64 + k*16 + f = M*16 + f  (since 64 = 4*16).
    const int col = lane & 15;
    const int rbase = (lane < 16) ? 0 : 8;
    float* accp = acc + (size_t)s * ACC_STRIDE + rbase * N_FEAT + col;

    // Rows rbase+0..3 are always valid (M = 0..3 or 8..11): no predication.
#pragma unroll
    for (int j = 0; j < 4; ++j) {
        unsafeAtomicAdd(accp + j * N_FEAT, D[j]);   // global_atomic_add_f32
    }
    // Rows 4..7 exist only for lanes 0-15 (rbase == 0): one EXEC toggle total.
    if (lane < 16) {
#pragma unroll
        for (int j = 4; j < 8; ++j) {
            unsafeAtomicAdd(accp + j * N_FEAT, D[j]);
        }
    }
}

// ---------------------------------------------------------------------------
// Kernel 3: _pos_to_rep remap. One thread per (node, col) pair; reads x,y,z
// from the accumulator (== imag half of d_out), writes real half + imag half
// in place. Each thread rewrites only addresses it itself read -> race-free.
// ---------------------------------------------------------------------------
__global__ void pos_to_rep_kernel(float* __restrict__ out,
                                  float* __restrict__ acc) {
    const int i = blockIdx.x * blockDim.x + threadIdx.x;
    const int total = N_NODES * 64;
    if (i >= total) return;
    const int n = i >> 6;        // node
    const int f = i & 63;        // flattened (basis, feat) column

    float* v  = acc + (size_t)n * ACC_STRIDE;   // node_vecs[n]
    const float x = v[f];
    const float y = v[64 + f];
    const float z = v[128 + f];

    float* re = out + (size_t)n * ACC_STRIDE;   // real block, node n
    re[f]        =  INV_SQRT2_C * x;
    re[64 + f]   =  z;
    re[128 + f]  = -INV_SQRT2_C * x;

    v[f]         = -INV_SQRT2_C * y;            // imag block (in place)
    v[64 + f]    =  0.0f;
    v[128 + f]   = -INV_SQRT2_C * y;
}

// ---------------------------------------------------------------------------
extern "C" void kernel_launch(void* const* d_in, const int* in_sizes, int n_in,
                              void* d_out, int out_size, void* d_ws, size_t ws_size,
                              hipStream_t stream) {
    const float* pos           = (const float*)d_in[0];
    const float* node_features = (const float*)d_in[1];
    const int*   edge_idx      = (const int*)d_in[2];

    float* out = (float*)d_out;
    // Accumulator lives in the imag half of d_out (exactly N_NODES*192 floats).
    float* acc = out + (size_t)N_NODES * ACC_STRIDE;

    (void)d_ws; (void)ws_size; (void)in_sizes; (void)n_in; (void)out_size;

    const int acc_elems = N_NODES * ACC_STRIDE;
    zero_acc_kernel<<<(acc_elems + 255) / 256, 256, 0, stream>>>(acc, acc_elems);

    // 8 waves (edges) per 256-thread block; 800000 / 8 = 100000 blocks exactly,
    // so every wave has EXEC all-ones at the WMMA (ISA requirement).
    edge_outer_wmma_kernel<<<N_EDGES / 8, 256, 0, stream>>>(pos, node_features,
                                                            edge_idx, acc);

    const int rep_threads = N_NODES * 64;
    pos_to_rep_kernel<<<(rep_threads + 255) / 256, 256, 0, stream>>>(out, acc);
}